// BaseModel_29162827940263
// MI455X (gfx1250) — compile-verified
//
#include <hip/hip_runtime.h>

typedef __attribute__((ext_vector_type(2))) float v2f;
typedef __attribute__((ext_vector_type(8))) float v8f;

constexpr int K      = 32;
constexpr int Bn     = 512;
constexpr int Tn     = 2048;
constexpr int START  = K - 2;
constexpr int STOP   = K - 1;
constexpr int CHUNKS = 8;
constexpr int CL     = Tn / CHUNKS;   // 256 steps per chunk
constexpr int WPB    = 4;             // waves per block

__device__ __forceinline__ float wave_max(float v) {
  #pragma unroll
  for (int off = 16; off > 0; off >>= 1) v = fmaxf(v, __shfl_xor(v, off, 32));
  return v;
}
__device__ __forceinline__ float wave_sum(float v) {
  #pragma unroll
  for (int off = 16; off > 0; off >>= 1) v += __shfl_xor(v, off, 32);
  return v;
}

// ---------------------------------------------------------------------------
// Phase 1: one wave per (batch b, chunk c). Maintains P = exp-domain transfer
// matrix (32x32 f32) for steps [c*CL, min((c+1)*CL, len)) as four 16x16 WMMA
// tiles, renormalized each step (scale accumulated in log space).
// Also fuses the gold-score partial sums (emission gather + transition path).
// ---------------------------------------------------------------------------
__global__ __launch_bounds__(128) void crf_phase1(
    const float* __restrict__ feats, const float* __restrict__ trans,
    const int* __restrict__ tags, const int* __restrict__ lens,
    float* __restrict__ chunkP, float* __restrict__ chunkScale,
    float* __restrict__ snapP, float* __restrict__ snapScale,
    float* __restrict__ goldPart) {
  __shared__ float s_trans[K * K];
  __shared__ float s_etrans[K * K];
  __shared__ float s_p[WPB][K * K];
  __shared__ float s_emit[WPB][K];

  for (int i = threadIdx.x; i < K * K; i += blockDim.x) {
    float tv = trans[i];
    s_trans[i]  = tv;
    s_etrans[i] = __expf(tv);   // exp(NEG) underflows to 0 == forbidden edge
  }
  __syncthreads();

  const int wslot = threadIdx.x >> 5;
  const int lane  = threadIdx.x & 31;
  const int wave  = blockIdx.x * WPB + wslot;
  const int b     = wave / CHUNKS;
  const int c     = wave % CHUNKS;
  const int len   = lens[b];

  int steps = len - c * CL;
  if (steps > CL) steps = CL;
  if (steps <= 0) {                       // chunk entirely past the sequence
    if (lane == 0) goldPart[b * CHUNKS + c] = 0.0f;
    return;
  }

  float* pw = s_p[wslot];
  #pragma unroll
  for (int r = 0; r < K; ++r) pw[r * K + lane] = (r == lane) ? 1.0f : 0.0f;

  const int half = lane >> 4;             // 0: lanes 0-15, 1: lanes 16-31
  const int m    = lane & 15;

  // A-operand fragments of E = exp(trans): tile row-half i, k-slice k.
  // 16x4 f32 A layout: VGPR0 = col (4k + 2*half), VGPR1 = col+1, row = m.
  v2f af[2][8];
  #pragma unroll
  for (int i = 0; i < 2; ++i)
    #pragma unroll
    for (int k = 0; k < 8; ++k) {
      const int col = k * 4 + half * 2;
      af[i][k][0] = s_etrans[(i * 16 + m) * K + col];
      af[i][k][1] = s_etrans[(i * 16 + m) * K + col + 1];
    }

  float logscale = 0.0f;
  float goldE = 0.0f;   // emission gold partial (lane-distributed)
  float goldT = 0.0f;   // transition gold partial (lane 0 only)

  const size_t fbase = (size_t)b * Tn * K;
  for (int tt = 0; tt < steps; ++tt) {
    const int t = c * CL + tt;
    const float e = feats[fbase + (size_t)t * K + lane];
    if (tt + 8 < steps)
      __builtin_prefetch(&feats[fbase + (size_t)(t + 8) * K + lane], 0, 0);
    s_emit[wslot][lane] = __expf(e);

    const int tag = tags[b * Tn + t];     // uniform broadcast load
    if (lane == tag) goldE += e;          // t < len always holds in this loop
    if (lane == 0) {
      const int prev = (t == 0) ? START : tags[b * Tn + t - 1];
      goldT += s_trans[tag * K + prev];
      if (t + 1 == len) goldT += s_trans[STOP * K + tag];
    }
    asm volatile("s_wait_dscnt 0" ::: "memory");

    // S = E * P  via 32 x V_WMMA_F32_16X16X4_F32
    v8f acc[2][2];
    #pragma unroll
    for (int i = 0; i < 2; ++i)
      #pragma unroll
      for (int j = 0; j < 2; ++j)
        #pragma unroll
        for (int r = 0; r < 8; ++r) acc[i][j][r] = 0.0f;

    #pragma unroll
    for (int k = 0; k < 8; ++k) {
      const int row0 = k * 4 + half * 2;  // B-operand 4x16: rows of P slice
      v2f b0, b1;
      b0[0] = pw[row0 * K + m];           b0[1] = pw[(row0 + 1) * K + m];
      b1[0] = pw[row0 * K + m + 16];      b1[1] = pw[(row0 + 1) * K + m + 16];
      #pragma unroll
      for (int i = 0; i < 2; ++i) {
        acc[i][0] = __builtin_amdgcn_wmma_f32_16x16x4_f32(
            false, af[i][k], false, b0, (short)0, acc[i][0], false, false);
        acc[i][1] = __builtin_amdgcn_wmma_f32_16x16x4_f32(
            false, af[i][k], false, b1, (short)0, acc[i][1], false, false);
      }
    }

    // Row-scale by exp(emit) (C/D layout: VGPR r holds row i*16 + 8*half + r),
    // then renormalize by the matrix max; track scale in log space.
    float mx = 0.0f;
    #pragma unroll
    for (int i = 0; i < 2; ++i)
      #pragma unroll
      for (int r = 0; r < 8; ++r) {
        const int row = i * 16 + half * 8 + r;
        const float sc = s_emit[wslot][row];
        const float v0 = acc[i][0][r] * sc;
        const float v1 = acc[i][1][r] * sc;
        acc[i][0][r] = v0;
        acc[i][1][r] = v1;
        mx = fmaxf(mx, fmaxf(v0, v1));
      }
    mx = wave_max(mx);
    const float inv = 1.0f / mx;
    logscale += __logf(mx);

    const bool snap = (t + 1 == len);     // wave-uniform
    #pragma unroll
    for (int i = 0; i < 2; ++i)
      #pragma unroll
      for (int r = 0; r < 8; ++r) {
        const int row = i * 16 + half * 8 + r;
        const float v0 = acc[i][0][r] * inv;
        const float v1 = acc[i][1][r] * inv;
        pw[row * K + m]      = v0;
        pw[row * K + m + 16] = v1;
        if (snap) {
          snapP[(size_t)b * K * K + row * K + m]      = v0;
          snapP[(size_t)b * K * K + row * K + m + 16] = v1;
        }
      }
    if (snap && lane == 0) snapScale[b] = logscale;
  }

  asm volatile("s_wait_dscnt 0" ::: "memory");
  if (steps == CL) {                      // full chunk: publish transfer matrix
    float* outP = chunkP + (size_t)(b * CHUNKS + c) * K * K;
    #pragma unroll
    for (int r = 0; r < K; ++r) outP[r * K + lane] = pw[r * K + lane];
    if (lane == 0) chunkScale[b * CHUNKS + c] = logscale;
  }
  const float ge = wave_sum(goldE);
  if (lane == 0) goldPart[b * CHUNKS + c] = ge + goldT;
}

// ---------------------------------------------------------------------------
// Phase 2: one wave per batch. Compose full chunks, then the at-len snapshot,
// in the scaled exp domain; finish forward score; subtract gold partials.
// ---------------------------------------------------------------------------
__global__ __launch_bounds__(128) void crf_phase2(
    const float* __restrict__ trans, const int* __restrict__ lens,
    const float* __restrict__ chunkP, const float* __restrict__ chunkScale,
    const float* __restrict__ snapP, const float* __restrict__ snapScale,
    const float* __restrict__ goldPart, float* __restrict__ perB) {
  const int lane = threadIdx.x & 31;
  const int b    = blockIdx.x * WPB + (threadIdx.x >> 5);
  if (b >= Bn) return;
  const int len  = lens[b];
  const int cLen = (len - 1) / CL;        // full chunks before snapshot chunk

  float p     = (lane == START) ? 1.0f : 0.0f;  // exp(init - 0)
  float scale = 0.0f;

  for (int cc = 0; cc < cLen; ++cc) {
    const float* M = chunkP + (size_t)(b * CHUNKS + cc) * K * K;
    float s = 0.0f;
    #pragma unroll 4
    for (int j = 0; j < K; ++j) s = fmaf(M[lane * K + j], __shfl(p, j, 32), s);
    const float mx = wave_max(s);
    p = s / mx;
    scale += chunkScale[b * CHUNKS + cc] + __logf(mx);
  }
  {
    const float* M = snapP + (size_t)b * K * K;
    float s = 0.0f;
    #pragma unroll 4
    for (int j = 0; j < K; ++j) s = fmaf(M[lane * K + j], __shfl(p, j, 32), s);
    const float mx = wave_max(s);
    p = s / mx;
    scale += snapScale[b] + __logf(mx);
  }

  float d = p * __expf(trans[STOP * K + lane]);
  d = wave_sum(d);
  if (lane == 0) {
    const float fwd = scale + __logf(d);
    float gold = 0.0f;
    #pragma unroll
    for (int cc = 0; cc < CHUNKS; ++cc) gold += goldPart[b * CHUNKS + cc];
    perB[b] = fwd - gold;
  }
}

// ---------------------------------------------------------------------------
// Phase 3: mean over the 512 per-batch values.
// ---------------------------------------------------------------------------
__global__ __launch_bounds__(256) void crf_phase3(const float* __restrict__ perB,
                                                  float* __restrict__ out) {
  __shared__ float red[256];
  red[threadIdx.x] = perB[threadIdx.x] + perB[threadIdx.x + 256];
  __syncthreads();
  for (int off = 128; off > 0; off >>= 1) {
    if (threadIdx.x < off) red[threadIdx.x] += red[threadIdx.x + off];
    __syncthreads();
  }
  if (threadIdx.x == 0) out[0] = red[0] * (1.0f / Bn);
}

extern "C" void kernel_launch(void* const* d_in, const int* in_sizes, int n_in,
                              void* d_out, int out_size, void* d_ws, size_t ws_size,
                              hipStream_t stream) {
  (void)in_sizes; (void)n_in; (void)out_size; (void)ws_size;
  const float* feats = (const float*)d_in[0];   // [B,T,K] f32
  const float* trans = (const float*)d_in[1];   // [K,K]   f32
  const int*   tags  = (const int*)d_in[2];     // [B,T]   i32
  const int*   lens  = (const int*)d_in[3];     // [B]     i32

  float* ws = (float*)d_ws;
  float* chunkP     = ws;                         // B*CHUNKS*K*K = 4,194,304 f
  float* snapP      = chunkP + (size_t)Bn * CHUNKS * K * K;   // B*K*K
  float* chunkScale = snapP + (size_t)Bn * K * K;             // B*CHUNKS
  float* snapScale  = chunkScale + Bn * CHUNKS;               // B
  float* goldPart   = snapScale + Bn;                         // B*CHUNKS
  float* perB       = goldPart + Bn * CHUNKS;                 // B

  crf_phase1<<<(Bn * CHUNKS) / WPB, 32 * WPB, 0, stream>>>(
      feats, trans, tags, lens, chunkP, chunkScale, snapP, snapScale, goldPart);
  crf_phase2<<<Bn / WPB, 32 * WPB, 0, stream>>>(
      trans, lens, chunkP, chunkScale, snapP, snapScale, goldPart, perB);
  crf_phase3<<<1, 256, 0, stream>>>(perB, (float*)d_out);
}